// RWKV_RNN_79766132621506
// MI455X (gfx1250) — compile-verified
//
#include <hip/hip_runtime.h>
#include <hip/hip_bf16.h>

#define D_DIM 1024
#define F_DIM 4096
#define L_LAYERS 24
#define V_DIM 50277
#define LN_EPS 1e-5f

typedef __attribute__((ext_vector_type(2))) float v2f;
typedef __attribute__((ext_vector_type(8))) float v8f;

__device__ __forceinline__ float sigmoidf(float v) {
    return 1.0f / (1.0f + __expf(-v));
}

// ---------------------------------------------------------------------------
// Block-wide sum + sumsq reduction (256 threads).
// ---------------------------------------------------------------------------
__device__ __forceinline__ void block_sums(float v, float vsq, float* red,
                                           float& out_sum, float& out_sq) {
    const int tid = threadIdx.x;
    red[tid] = v;
    red[256 + tid] = vsq;
    __syncthreads();
    for (int s = 128; s > 0; s >>= 1) {
        if (tid < s) {
            red[tid] += red[tid + s];
            red[256 + tid] += red[256 + tid + s];
        }
        __syncthreads();
    }
    out_sum = red[0];
    out_sq = red[256];
    __syncthreads();
}

// ---------------------------------------------------------------------------
// x = LN(emb[token]) with ln0
// ---------------------------------------------------------------------------
__global__ __launch_bounds__(256) void embed_ln0_kernel(
    const int* __restrict__ token, const float* __restrict__ emb,
    const float* __restrict__ w, const float* __restrict__ b,
    float* __restrict__ xo) {
    __shared__ float red[512];
    const float* e = emb + (size_t)token[0] * D_DIM;
    float v[4];
    float s = 0.f, sq = 0.f;
#pragma unroll
    for (int j = 0; j < 4; ++j) {
        int i = threadIdx.x + 256 * j;
        v[j] = e[i];
        s += v[j];
        sq += v[j] * v[j];
    }
    float S, SQ;
    block_sums(s, sq, red, S, SQ);
    float mu = S * (1.0f / D_DIM);
    float var = SQ * (1.0f / D_DIM) - mu * mu;
    float rstd = rsqrtf(var + LN_EPS);
#pragma unroll
    for (int j = 0; j < 4; ++j) {
        int i = threadIdx.x + 256 * j;
        xo[i] = (v[j] - mu) * rstd * w[i] + b[i];
    }
}

// ---------------------------------------------------------------------------
// Plain LN (for ln_out)
// ---------------------------------------------------------------------------
__global__ __launch_bounds__(256) void ln_kernel(
    const float* __restrict__ xin, const float* __restrict__ w,
    const float* __restrict__ b, float* __restrict__ xo) {
    __shared__ float red[512];
    float v[4];
    float s = 0.f, sq = 0.f;
#pragma unroll
    for (int j = 0; j < 4; ++j) {
        int i = threadIdx.x + 256 * j;
        v[j] = xin[i];
        s += v[j];
        sq += v[j] * v[j];
    }
    float S, SQ;
    block_sums(s, sq, red, S, SQ);
    float mu = S * (1.0f / D_DIM);
    float var = SQ * (1.0f / D_DIM) - mu * mu;
    float rstd = rsqrtf(var + LN_EPS);
#pragma unroll
    for (int j = 0; j < 4; ++j) {
        int i = threadIdx.x + 256 * j;
        xo[i] = (v[j] - mu) * rstd * w[i] + b[i];
    }
}

// ---------------------------------------------------------------------------
// xx = LN(x, l1) ; xk/xv/xr token-mix against state slot 1.
// xx also written to out_state slot 1.
// ---------------------------------------------------------------------------
__global__ __launch_bounds__(256) void ln_mix_att_kernel(
    const float* __restrict__ x, const float* __restrict__ w,
    const float* __restrict__ b, const float* __restrict__ tmk,
    const float* __restrict__ tmv, const float* __restrict__ tmr,
    const float* __restrict__ s1, float* __restrict__ xx_out,
    float* __restrict__ xk, float* __restrict__ xv, float* __restrict__ xr) {
    __shared__ float red[512];
    float v[4];
    float s = 0.f, sq = 0.f;
#pragma unroll
    for (int j = 0; j < 4; ++j) {
        int i = threadIdx.x + 256 * j;
        v[j] = x[i];
        s += v[j];
        sq += v[j] * v[j];
    }
    float S, SQ;
    block_sums(s, sq, red, S, SQ);
    float mu = S * (1.0f / D_DIM);
    float var = SQ * (1.0f / D_DIM) - mu * mu;
    float rstd = rsqrtf(var + LN_EPS);
#pragma unroll
    for (int j = 0; j < 4; ++j) {
        int i = threadIdx.x + 256 * j;
        float xx = (v[j] - mu) * rstd * w[i] + b[i];
        float sv = s1[i];
        xx_out[i] = xx;
        float mk = tmk[i], mv = tmv[i], mr = tmr[i];
        xk[i] = xx * mk + sv * (1.0f - mk);
        xv[i] = xx * mv + sv * (1.0f - mv);
        xr[i] = xx * mr + sv * (1.0f - mr);
    }
}

// ---------------------------------------------------------------------------
// xx2 = LN(x, l2) ; xk2/xr2 token-mix against state slot 0.
// xx2 also written to out_state slot 0.
// ---------------------------------------------------------------------------
__global__ __launch_bounds__(256) void ln_mix_ffn_kernel(
    const float* __restrict__ x, const float* __restrict__ w,
    const float* __restrict__ b, const float* __restrict__ tmk,
    const float* __restrict__ tmr, const float* __restrict__ s0,
    float* __restrict__ xx_out, float* __restrict__ xk,
    float* __restrict__ xr) {
    __shared__ float red[512];
    float v[4];
    float s = 0.f, sq = 0.f;
#pragma unroll
    for (int j = 0; j < 4; ++j) {
        int i = threadIdx.x + 256 * j;
        v[j] = x[i];
        s += v[j];
        sq += v[j] * v[j];
    }
    float S, SQ;
    block_sums(s, sq, red, S, SQ);
    float mu = S * (1.0f / D_DIM);
    float var = SQ * (1.0f / D_DIM) - mu * mu;
    float rstd = rsqrtf(var + LN_EPS);
#pragma unroll
    for (int j = 0; j < 4; ++j) {
        int i = threadIdx.x + 256 * j;
        float xx = (v[j] - mu) * rstd * w[i] + b[i];
        float sv = s0[i];
        xx_out[i] = xx;
        float mk = tmk[i], mr = tmr[i];
        xk[i] = xx * mk + sv * (1.0f - mk);
        xr[i] = xx * mr + sv * (1.0f - mr);
    }
}

// ---------------------------------------------------------------------------
// WKV recurrence (elementwise) + state slots 2/3/4 update.
// ---------------------------------------------------------------------------
__global__ __launch_bounds__(256) void wkv_kernel(
    const float* __restrict__ kbuf, const float* __restrict__ vbuf,
    const float* __restrict__ rpre, const float* __restrict__ tf,
    const float* __restrict__ td, const float* __restrict__ aa,
    const float* __restrict__ bb, const float* __restrict__ pp,
    float* __restrict__ rwkv, float* __restrict__ aa_o,
    float* __restrict__ bb_o, float* __restrict__ pp_o) {
    for (int i = threadIdx.x; i < D_DIM; i += 256) {
        float k = kbuf[i], v = vbuf[i];
        float A = aa[i], B = bb[i], P = pp[i];
        float ww = tf[i] + k;
        float p_ = fmaxf(P, ww);
        float e1 = __expf(P - p_);
        float e2 = __expf(ww - p_);
        float wkv = (e1 * A + e2 * v) / (e1 * B + e2);
        rwkv[i] = sigmoidf(rpre[i]) * wkv;
        float ww2 = P + td[i];
        float p2 = fmaxf(ww2, k);
        float e1b = __expf(ww2 - p2);
        float e2b = __expf(k - p2);
        aa_o[i] = e1b * A + e2b * v;
        bb_o[i] = e1b * B + e2b;
        pp_o[i] = p2;
    }
}

// ---------------------------------------------------------------------------
// WMMA-based matvec: out[r] = epi( dot(W[r,:], x) ), W row-major nrows x ncols.
//
// Block = 256 thr = 8 waves. Block owns a 16-row M-tile; each wave takes a
// K-chunk of ncols/8. Uses V_WMMA_F32_16X16X4_F32:
//   A tile (exact ISA layout): lane 0-15 -> M, VGPR0 holds K={0|2} per lane
//   half, VGPR1 holds K={1|3} -> one b64 load per lane per step.
//   B = x broadcast across all 16 columns (each column of D is the matvec).
// x is staged in LDS (b64 reads), partials reduced across waves via LDS.
// MODE: 0=store, 1=out+=d (residual), 2=relu(d)^2, 3=out+=sigmoid(gate)*d.
// ---------------------------------------------------------------------------
template <int MODE>
__global__ __launch_bounds__(256) void matvec_wmma_kernel(
    const float* __restrict__ W, const float* __restrict__ x,
    const float* __restrict__ gate, float* __restrict__ out, int nrows,
    int ncols) {
    __shared__ float xs[F_DIM];
    __shared__ float red[8][16];

    const int tid = threadIdx.x;
    const int wave = tid >> 5;
    const int lane = tid & 31;
    const int half = lane >> 4;
    const int rbase = blockIdx.x << 4;

    for (int i = tid; i < ncols; i += 256) xs[i] = x[i];
    __syncthreads();

    int row = rbase + (lane & 15);
    row = (row < nrows) ? row : (nrows - 1);  // clamp keeps EXEC uniform

    const int kw = ncols >> 3;  // per-wave K extent (>= 128, multiple of 64)
    const int k0w = wave * kw;
    const float* wp = W + (size_t)row * ncols + k0w + (half << 1);
    const float* bp = xs + k0w + (half << 1);

    v8f acc = {};
    for (int k = 0; k < kw; k += 64) {
        __builtin_prefetch(wp + 256, 0, 1);  // ~1KB ahead on this row stream
#pragma unroll
        for (int kk = 0; kk < 64; kk += 4) {
            v2f a = *(const v2f*)wp;  // W[row][k+2*half], W[row][k+2*half+1]
            v2f b;
            b.x = bp[0];  // x[k+2*half]   (B VGPR0: K=0 / K=2 per half)
            b.y = bp[1];  // x[k+2*half+1] (B VGPR1: K=1 / K=3 per half)
            acc = __builtin_amdgcn_wmma_f32_16x16x4_f32(
                false, a, false, b, (short)0, acc, false, false);
            wp += 4;
            bp += 4;
        }
    }

    // D layout: VGPR j, lanes 0-15 -> M=j, lanes 16-31 -> M=j+8 (all N equal).
    if ((lane & 15) == 0) {
#pragma unroll
        for (int j = 0; j < 8; ++j) red[wave][(half << 3) + j] = acc[j];
    }
    __syncthreads();

    if (tid < 16) {
        int r = rbase + tid;
        if (r < nrows) {
            float d = 0.f;
#pragma unroll
            for (int w2 = 0; w2 < 8; ++w2) d += red[w2][tid];
            if (MODE == 0) {
                out[r] = d;
            } else if (MODE == 1) {
                out[r] += d;
            } else if (MODE == 2) {
                float t = fmaxf(d, 0.0f);
                out[r] = t * t;
            } else {
                out[r] += sigmoidf(gate[r]) * d;
            }
        }
    }
}

// ---------------------------------------------------------------------------
// Launch: full RWKV token step as a stream-ordered kernel sequence.
// ---------------------------------------------------------------------------
extern "C" void kernel_launch(void* const* d_in, const int* in_sizes, int n_in,
                              void* d_out, int out_size, void* d_ws,
                              size_t ws_size, hipStream_t stream) {
    const int* token = (const int*)d_in[0];
    const float* state = (const float*)d_in[1];
    const float* emb = (const float*)d_in[2];
    const float* ln0w = (const float*)d_in[3];
    const float* ln0b = (const float*)d_in[4];
    const float* ln1w = (const float*)d_in[5];
    const float* ln1b = (const float*)d_in[6];
    const float* ln2w = (const float*)d_in[7];
    const float* ln2b = (const float*)d_in[8];
    const float* atmk = (const float*)d_in[9];
    const float* atmv = (const float*)d_in[10];
    const float* atmr = (const float*)d_in[11];
    const float* tf = (const float*)d_in[12];
    const float* td = (const float*)d_in[13];
    const float* akw = (const float*)d_in[14];
    const float* avw = (const float*)d_in[15];
    const float* arw = (const float*)d_in[16];
    const float* aow = (const float*)d_in[17];
    const float* ftmk = (const float*)d_in[18];
    const float* ftmr = (const float*)d_in[19];
    const float* fkw = (const float*)d_in[20];
    const float* fvw = (const float*)d_in[21];
    const float* frw = (const float*)d_in[22];
    const float* lnow = (const float*)d_in[23];
    const float* lnob = (const float*)d_in[24];
    const float* headw = (const float*)d_in[25];

    float* logits = (float*)d_out;            // V floats
    float* ostate = logits + V_DIM;           // L*5*D floats

    float* ws = (float*)d_ws;
    float* x = ws + 0 * D_DIM;
    float* xk = ws + 1 * D_DIM;
    float* xv = ws + 2 * D_DIM;
    float* xr = ws + 3 * D_DIM;
    float* kb = ws + 4 * D_DIM;
    float* vb = ws + 5 * D_DIM;
    float* rp = ws + 6 * D_DIM;
    float* rwkv = ws + 7 * D_DIM;
    float* xk2 = ws + 8 * D_DIM;
    float* xr2 = ws + 9 * D_DIM;
    float* r2p = ws + 10 * D_DIM;
    float* xo = ws + 11 * D_DIM;
    float* k2 = ws + 12 * D_DIM;  // F floats

    embed_ln0_kernel<<<1, 256, 0, stream>>>(token, emb, ln0w, ln0b, x);

    for (int l = 0; l < L_LAYERS; ++l) {
        const float* st = state + (size_t)l * 5 * D_DIM;
        float* ost = ostate + (size_t)l * 5 * D_DIM;
        const size_t dd = (size_t)l * D_DIM * D_DIM;
        const size_t fd = (size_t)l * F_DIM * D_DIM;
        const size_t ld = (size_t)l * D_DIM;

        ln_mix_att_kernel<<<1, 256, 0, stream>>>(
            x, ln1w + ld, ln1b + ld, atmk + ld, atmv + ld, atmr + ld,
            st + D_DIM, ost + D_DIM, xk, xv, xr);

        matvec_wmma_kernel<0><<<D_DIM / 16, 256, 0, stream>>>(
            akw + dd, xk, nullptr, kb, D_DIM, D_DIM);
        matvec_wmma_kernel<0><<<D_DIM / 16, 256, 0, stream>>>(
            avw + dd, xv, nullptr, vb, D_DIM, D_DIM);
        matvec_wmma_kernel<0><<<D_DIM / 16, 256, 0, stream>>>(
            arw + dd, xr, nullptr, rp, D_DIM, D_DIM);

        wkv_kernel<<<1, 256, 0, stream>>>(
            kb, vb, rp, tf + ld, td + ld, st + 2 * D_DIM, st + 3 * D_DIM,
            st + 4 * D_DIM, rwkv, ost + 2 * D_DIM, ost + 3 * D_DIM,
            ost + 4 * D_DIM);

        matvec_wmma_kernel<1><<<D_DIM / 16, 256, 0, stream>>>(
            aow + dd, rwkv, nullptr, x, D_DIM, D_DIM);  // x += ow @ (r*wkv)

        ln_mix_ffn_kernel<<<1, 256, 0, stream>>>(
            x, ln2w + ld, ln2b + ld, ftmk + ld, ftmr + ld, st, ost, xk2, xr2);

        matvec_wmma_kernel<0><<<D_DIM / 16, 256, 0, stream>>>(
            frw + dd, xr2, nullptr, r2p, D_DIM, D_DIM);
        matvec_wmma_kernel<2><<<F_DIM / 16, 256, 0, stream>>>(
            fkw + fd, xk2, nullptr, k2, F_DIM, D_DIM);  // relu(.)^2
        matvec_wmma_kernel<3><<<D_DIM / 16, 256, 0, stream>>>(
            fvw + fd, k2, r2p, x, D_DIM, F_DIM);  // x += sig(r2) * (fvw@k2)
    }

    ln_kernel<<<1, 256, 0, stream>>>(x, lnow, lnob, xo);
    matvec_wmma_kernel<0><<<(V_DIM + 15) / 16, 256, 0, stream>>>(
        headw, xo, nullptr, logits, V_DIM, D_DIM);
}